// GlobalWorkspace_87806311400115
// MI455X (gfx1250) — compile-verified
//
#include <hip/hip_runtime.h>
#include <hip/hip_bf16.h>

typedef __attribute__((ext_vector_type(2))) float v2f;
typedef __attribute__((ext_vector_type(8))) float v8f;
typedef unsigned long long u64;
typedef unsigned int u32;

#define T_STEPS 8192
#define N_NEUR  4096
#define ALPHA   0.01f
#define DECAY   0.99f
#define THR     0.58f

#define NSEG        16               // time segments (parallel scan blocks)
#define SEGLEN      (T_STEPS / NSEG) // 512
#define NCOLBLK     32               // column blocks of 128 cols (8 waves x 16 cols)
#define COLS_PER_BLOCK 128

// Sortable key: nmda >= 0 always, so float bits are monotone as u32.
// key = (bits << 32) | ~index  ->  u64 '>' == (value greater) or (equal, index lower)
struct Key2 { u64 a; u64 b; };   // a >= b (top-2, sorted)

__device__ __forceinline__ u64 makeKey(float v, u32 invcol) {
  return ((u64)__float_as_uint(v) << 32) | invcol;
}

// merge sorted pairs {a0>=a1} and {b0>=b1} -> top-2 of union (branch-free)
__device__ __forceinline__ void merge2(u64& a0, u64& a1, u64 b0, u64 b1) {
  const bool aw = a0 > b0;
  u64 lo = aw ? b0 : a0;        // loser of the firsts
  u64 c  = aw ? a1 : b1;        // winner's second
  a0 = aw ? a0 : b0;
  a1 = lo > c ? lo : c;
}

// 64-bit lane exchange via two DPP16 moves (CTRL: quad_perm / row_ror, compile-time)
template <int CTRL>
__device__ __forceinline__ u64 dpp64(u64 x) {
  int lo = (int)(u32)x;
  int hi = (int)(u32)(x >> 32);
  lo = __builtin_amdgcn_mov_dpp(lo, CTRL, 0xf, 0xf, true);
  hi = __builtin_amdgcn_mov_dpp(hi, CTRL, 0xf, 0xf, true);
  return ((u64)(u32)hi << 32) | (u32)lo;
}

__device__ __forceinline__ float pow99(int d) {
  float p = 1.0f;
  for (int i = 0; i < d; ++i) p *= DECAY;   // same rounding chain as repeated 0.99*x
  return p;
}

// ---------------------------------------------------------------------------
// Kernel 0: zero-fill the ignite mask (128 MiB) with 128-bit stores.
// ---------------------------------------------------------------------------
__global__ void gw_zero_kernel(float4* __restrict__ p, int n4) {
  int idx = blockIdx.x * blockDim.x + threadIdx.x;
  if (idx < n4) p[idx] = make_float4(0.f, 0.f, 0.f, 0.f);
}

// ---------------------------------------------------------------------------
// Kernel 1: per-segment local EMA final (zero initial state).
// ---------------------------------------------------------------------------
__global__ __launch_bounds__(256)
void gw_segfinal_kernel(const float* __restrict__ spikes, float* __restrict__ lf) {
  const int seg = blockIdx.x;
  const int col = blockIdx.y * 256 + threadIdx.x;
  const float* p = spikes + (size_t)(seg * SEGLEN) * N_NEUR + col;
  float acc = 0.0f;
  for (int i = 0; i < SEGLEN; ++i) {
    acc = DECAY * acc + ALPHA * p[(size_t)i * N_NEUR];
  }
  lf[(size_t)seg * N_NEUR + col] = acc;
}

// ---------------------------------------------------------------------------
// Kernel 2: compose segment carries.  carr[g][col] = nmda state entering seg g.
// ---------------------------------------------------------------------------
__global__ __launch_bounds__(256)
void gw_carry_kernel(const float* __restrict__ lf, const float* __restrict__ nmda0,
                     float* __restrict__ carr) {
  const int col = blockIdx.x * 256 + threadIdx.x;
  const float d512 = pow99(SEGLEN);
  float c = nmda0[col];
  for (int g = 0; g < NSEG; ++g) {
    carr[(size_t)g * N_NEUR + col] = c;
    c = d512 * c + lf[(size_t)g * N_NEUR + col];
  }
}

// ---------------------------------------------------------------------------
// Kernel 3: WMMA tile scan + fused per-row top-2 (DPP gossip reduction).
// grid = (NCOLBLK, NSEG); block = 256 (8 waves, 16 cols each).
// Per 16-step tile: nmda = W @ S  (4x V_WMMA_F32_16X16X4_F32)  + 0.99^{i+1}*carry
// ---------------------------------------------------------------------------
__global__ __launch_bounds__(256)
void gw_scan_wmma_kernel(const float* __restrict__ spikes,
                         const float* __restrict__ carr,
                         float* __restrict__ nmda_final,
                         Key2* __restrict__ partials) {
  const int tid   = threadIdx.x;
  const int wave  = tid >> 5;
  const int lane  = tid & 31;
  const int m     = lane & 15;
  const bool hi   = lane >= 16;
  const int seg   = blockIdx.y;
  const int col0  = blockIdx.x * COLS_PER_BLOCK + wave * 16;
  const int mycol = col0 + m;
  const u32 invcol = ~(u32)mycol;

  // A = W (lower-triangular 16x16), split into four 16x4 f32 A-operands.
  v2f A[4];
#pragma unroll
  for (int kb = 0; kb < 4; ++kb) {
    int j0 = kb * 4 + (hi ? 2 : 0);
    int j1 = j0 + 1;
    A[kb].x = (j0 <= m) ? ALPHA * pow99(m - j0) : 0.0f;
    A[kb].y = (j1 <= m) ? ALPHA * pow99(m - j1) : 0.0f;
  }
  // carry decay per C/D VGPR row: row = r + (hi?8:0), factor 0.99^{row+1}
  float pw[8];
#pragma unroll
  for (int r = 0; r < 8; ++r) pw[r] = pow99(r + (hi ? 8 : 0) + 1);

  const int tbeg = seg * SEGLEN;
  float cc = carr[(size_t)seg * N_NEUR + mycol];   // carry entering this segment

  // strided base pointer: per-kb rows sit at constant immediate offsets
  const float* sp = spikes + (size_t)(tbeg + (hi ? 2 : 0)) * N_NEUR + mycol;

  __shared__ Key2 lds[8][16];

  for (int tt = tbeg; tt < tbeg + SEGLEN; tt += 16) {
    v8f c = {};
#pragma unroll
    for (int kb = 0; kb < 4; ++kb) {
      v2f b;
      b.x = sp[(size_t)(kb * 4)     * N_NEUR];   // imm offset kb*64KiB
      b.y = sp[(size_t)(kb * 4 + 1) * N_NEUR];   // imm offset kb*64KiB+16KiB
      c = __builtin_amdgcn_wmma_f32_16x16x4_f32(
              /*neg_a=*/false, A[kb], /*neg_b=*/false, b,
              /*c_mod=*/(short)0, c, /*reuse_a=*/false, /*reuse_b=*/false);
    }
    sp += (size_t)16 * N_NEUR;

    // nmda[row] = (W@S)[row] + 0.99^{row+1} * carry
    float nv[8];
#pragma unroll
    for (int r = 0; r < 8; ++r) nv[r] = c[r] + pw[r] * cc;

    // new carry = nmda at row 15 (VGPR7, lanes 16-31), broadcast per column
    cc = __shfl(nv[7], 16 + m, 32);

    // per-row top-2 over 16 lanes: quad_perm(x1), quad_perm(x2), row_ror:4, row_ror:8
    Key2 recs[8];
#pragma unroll
    for (int r = 0; r < 8; ++r) {
      u64 k = makeKey(nv[r], invcol);
      // step 1 (distance 1): merge two singletons
      u64 o  = dpp64<0xB1>(k);             // quad_perm [1,0,3,2]
      u64 k0 = k > o ? k : o;
      u64 k1 = k > o ? o : k;
      // step 2 (distance 2): quad -> top2 of its quad
      merge2(k0, k1, dpp64<0x4E>(k0), dpp64<0x4E>(k1));   // quad_perm [2,3,0,1]
      // step 3: rotate by 4 within 16-lane row -> top2 of two quads
      merge2(k0, k1, dpp64<0x124>(k0), dpp64<0x124>(k1)); // row_ror:4
      // step 4: rotate by 8 -> top2 of all four quads (whole 16-group)
      merge2(k0, k1, dpp64<0x128>(k0), dpp64<0x128>(k1)); // row_ror:8
      recs[r] = Key2{k0, k1};
    }
    if (m == 0) {                          // single guarded 8-store DS clause
#pragma unroll
      for (int r = 0; r < 8; ++r) lds[wave][r + (hi ? 8 : 0)] = recs[r];
    }
    __syncthreads();
    if (tid < 16) {                 // merge 8 waves -> block partial for row tt+tid
      Key2 rec = lds[0][tid];
#pragma unroll
      for (int w = 1; w < 8; ++w) {
        Key2 o2 = lds[w][tid];
        merge2(rec.a, rec.b, o2.a, o2.b);
      }
      partials[(size_t)(tt + tid) * NCOLBLK + blockIdx.x] = rec;
    }
    __syncthreads();
  }

  if (seg == NSEG - 1 && !hi) nmda_final[mycol] = cc;
}

// ---------------------------------------------------------------------------
// Kernel 4: final merge across 32 column blocks, ignition test, sparse scatter.
// ---------------------------------------------------------------------------
__global__ __launch_bounds__(256)
void gw_combine_kernel(const Key2* __restrict__ partials,
                       float* __restrict__ mask, float* __restrict__ coverage) {
  int t = blockIdx.x * blockDim.x + threadIdx.x;
  if (t >= T_STEPS) return;
  Key2 rec = partials[(size_t)t * NCOLBLK];
  for (int b = 1; b < NCOLBLK; ++b) {
    Key2 o = partials[(size_t)t * NCOLBLK + b];
    merge2(rec.a, rec.b, o.a, o.b);
  }
  const float vmax = __uint_as_float((u32)(rec.a >> 32));
  const int   i0   = (int)(~(u32)rec.a);
  const int   i1   = (int)(~(u32)rec.b);
  const bool  ign  = vmax >= THR;                 // max(nmda) >= ignite_thr
  coverage[t] = ign ? (2.0f / 4096.0f) : 0.0f;    // k/N exactly
  if (ign) {                                      // 0.85 scaling preserves order
    mask[(size_t)t * N_NEUR + i0] = 1.0f;
    mask[(size_t)t * N_NEUR + i1] = 1.0f;
  }
}

// ---------------------------------------------------------------------------
extern "C" void kernel_launch(void* const* d_in, const int* in_sizes, int n_in,
                              void* d_out, int out_size, void* d_ws, size_t ws_size,
                              hipStream_t stream) {
  const float* spikes = (const float*)d_in[0];   // [8192, 4096]
  const float* nmda0  = (const float*)d_in[1];   // [4096]

  float* mask     = (float*)d_out;                       // [T, N]
  float* coverage = mask + (size_t)T_STEPS * N_NEUR;     // [T]
  float* nfinal   = coverage + T_STEPS;                  // [N]

  // workspace layout
  Key2*  partials = (Key2*)d_ws;                                       // 4 MiB
  float* lf       = (float*)((char*)d_ws + (size_t)T_STEPS * NCOLBLK * sizeof(Key2));
  float* carr     = lf + (size_t)NSEG * N_NEUR;                        // 256 KiB each

  const int n4 = (T_STEPS * N_NEUR) / 4;
  gw_zero_kernel<<<(n4 + 255) / 256, 256, 0, stream>>>((float4*)mask, n4);

  gw_segfinal_kernel<<<dim3(NSEG, N_NEUR / 256), 256, 0, stream>>>(spikes, lf);

  gw_carry_kernel<<<N_NEUR / 256, 256, 0, stream>>>(lf, nmda0, carr);

  gw_scan_wmma_kernel<<<dim3(NCOLBLK, NSEG), 256, 0, stream>>>(spikes, carr, nfinal, partials);

  gw_combine_kernel<<<(T_STEPS + 255) / 256, 256, 0, stream>>>(partials, mask, coverage);
}